// QFDloss_36344013258960
// MI455X (gfx1250) — compile-verified
//
#include <hip/hip_runtime.h>
#include <hip/hip_bf16.h>
#include <math.h>

typedef float v2f __attribute__((ext_vector_type(2)));
typedef float v8f __attribute__((ext_vector_type(8)));

#define NBLOCKS 1024
#define BLOCK   256          // 8 waves (wave32)
#define NWAVES  (BLOCK / 32)
#define ROWS    64           // rows of diff per LDS chunk
#define DLEN    85
#define STRIDE  104          // LDS row stride (floats); 2*104 mod 64 = 16 -> halves hit disjoint banks
#define NTILE   6            // 6 column tiles of 16 (96 padded cols)
#define NPAIRS  21           // upper triangle of 6x6 tile grid
#define TPW     3            // tiles per wave (8 waves x 3 = 24 >= 21, rest are zero-weighted dummies)

__global__ __launch_bounds__(BLOCK) void qfd_gram_kernel(
    const float* __restrict__ in, const float* __restrict__ tgt,
    float* __restrict__ partial, int nchunks)
{
    __shared__ float lds[ROWS * STRIDE];
    __shared__ float redbuf[NWAVES];

    const int tid  = threadIdx.x;
    const int lane = tid & 31;
    const int wave = __builtin_amdgcn_readfirstlane(tid >> 5);
    const int half = lane >> 4;   // half-wave selects the K pair
    const int l15  = lane & 15;

    // Zero pad columns [85, STRIDE) once; fill loop never touches them.
    for (int i = tid; i < ROWS * (STRIDE - DLEN); i += BLOCK) {
        int r = i / (STRIDE - DLEN);
        int c = DLEN + (i - r * (STRIDE - DLEN));
        lds[r * STRIDE + c] = 0.0f;
    }

    // Decode this wave's 3 tile pairs (p = wave + 8*t). p >= 21 -> dummy tile (0,0), weight 0.
    int   ti_[TPW], tj_[TPW], ca_[TPW], cb_[TPW];
    float mult_[TPW];
#pragma unroll
    for (int t = 0; t < TPW; ++t) {
        int p   = wave + 8 * t;
        int rem = (p < NPAIRS) ? p : 0;
        int ti = 0, len = NTILE;
        while (rem >= len) { rem -= len; ++ti; --len; }
        int tj = ti + rem;
        ti_[t] = ti;
        tj_[t] = tj;
        ca_[t] = ti * 16 + l15;   // A-fragment LDS column (M = lane)
        cb_[t] = tj * 16 + l15;   // B-fragment LDS column (N = lane)
        mult_[t] = (p >= NPAIRS) ? 0.0f : (ti == tj ? 1.0f : 2.0f);
    }

    v8f acc[TPW];
#pragma unroll
    for (int t = 0; t < TPW; ++t) acc[t] = (v8f){0.f,0.f,0.f,0.f,0.f,0.f,0.f,0.f};

    for (int chunk = blockIdx.x; chunk < nchunks; chunk += gridDim.x) {
        __syncthreads();   // previous chunk's LDS readers done before overwrite

        // Stream 64 rows x 85 cols of |in - tgt| into padded LDS (float4 global loads).
        const float4* in4 = (const float4*)(in  + (size_t)chunk * (ROWS * DLEN));
        const float4* tg4 = (const float4*)(tgt + (size_t)chunk * (ROWS * DLEN));
        for (int i = tid; i < (ROWS * DLEN) / 4; i += BLOCK) {
            float4 a = in4[i];
            float4 b = tg4[i];
            float dv[4] = { fabsf(a.x - b.x), fabsf(a.y - b.y),
                            fabsf(a.z - b.z), fabsf(a.w - b.w) };
            int idx = i * 4;
            int r = idx / DLEN;
            int c = idx - r * DLEN;
#pragma unroll
            for (int e = 0; e < 4; ++e) {
                lds[r * STRIDE + c] = dv[e];
                if (++c == DLEN) { c = 0; ++r; }
            }
        }
        __syncthreads();

        // Branch-free Gram accumulation: per K-step, issue ALL fragment loads first,
        // then the three independent in-place WMMA chains (lets LDS latency overlap).
#pragma unroll 4
        for (int k = 0; k < ROWS; k += 4) {
            const float* row0 = &lds[(k + 2 * half) * STRIDE];
            const float* row1 = row0 + STRIDE;
            v2f Af[TPW], Bf[TPW];
#pragma unroll
            for (int t = 0; t < TPW; ++t) {
                Af[t] = (v2f){ row0[ca_[t]], row1[ca_[t]] };   // A 16x4 f32 frag
                Bf[t] = (v2f){ row0[cb_[t]], row1[cb_[t]] };   // B 4x16 f32 frag
            }
#pragma unroll
            for (int t = 0; t < TPW; ++t) {
                acc[t] = __builtin_amdgcn_wmma_f32_16x16x4_f32(
                    false, Af[t], false, Bf[t], (short)0, acc[t], false, false);
            }
        }
    }

    // Weighted contraction <A, G>, analytic A[i][j] = 1 - |i-j|/85 (0 in padding / dummy tiles).
    float local = 0.0f;
#pragma unroll
    for (int t = 0; t < TPW; ++t) {
        int jcol = tj_[t] * 16 + l15;
#pragma unroll
        for (int v = 0; v < 8; ++v) {
            int irow = ti_[t] * 16 + v + 8 * half;   // C/D layout: VGPR v -> M = v + 8*half
            float w = 0.0f;
            if (irow < DLEN && jcol < DLEN)
                w = 1.0f - fabsf((float)(irow - jcol)) * (1.0f / 85.0f);
            local += mult_[t] * w * acc[t][v];
        }
    }

    // Fixed-tree wave reduction, then fixed-order cross-wave sum (deterministic).
    for (int off = 16; off > 0; off >>= 1)
        local += __shfl_down(local, off, 32);
    if (lane == 0) redbuf[wave] = local;
    __syncthreads();
    if (tid == 0) {
        float s = 0.0f;
        for (int wv = 0; wv < NWAVES; ++wv) s += redbuf[wv];
        partial[blockIdx.x] = s;
    }
}

__global__ __launch_bounds__(256) void qfd_reduce_kernel(
    const float* __restrict__ partial, float* __restrict__ out)
{
    __shared__ float sb[256];
    int tid = threadIdx.x;
    float s = 0.0f;
    for (int i = tid; i < NBLOCKS; i += 256) s += partial[i];
    sb[tid] = s;
    __syncthreads();
    for (int st = 128; st > 0; st >>= 1) {
        if (tid < st) sb[tid] += sb[tid + st];
        __syncthreads();
    }
    if (tid == 0) out[0] = 0.1f * sb[0];
}

extern "C" void kernel_launch(void* const* d_in, const int* in_sizes, int n_in,
                              void* d_out, int out_size, void* d_ws, size_t ws_size,
                              hipStream_t stream) {
    const float* in  = (const float*)d_in[0];
    const float* tgt = (const float*)d_in[1];
    float* out = (float*)d_out;
    float* ws  = (float*)d_ws;   // NBLOCKS floats of partial sums

    int n = in_sizes[0];         // B * 85
    int B = n / DLEN;            // 262144
    int nchunks = B / ROWS;      // 4096 (B divisible by 64)

    qfd_gram_kernel<<<NBLOCKS, BLOCK, 0, stream>>>(in, tgt, ws, nchunks);
    qfd_reduce_kernel<<<1, 256, 0, stream>>>(ws, out);
}